// Model_23141283791190
// MI455X (gfx1250) — compile-verified
//
#include <hip/hip_runtime.h>
#include <stdint.h>

#define N_SAMPLES   32768
#define KERNEL_SIZE 4096
#define N_ITERS     32
#define D_SIZE      2048
#define N_FRAMES    512
#define STEP        64
#define BATCH       32

#define TCHUNK      2048                   // samples per output block
#define NCHUNK      (N_SAMPLES / TCHUNK)   // 16
#define BLK         256
#define VPT         (TCHUNK / BLK)         // 8 samples per thread

#define KSLICES     8
#define KSLICE_LEN  (KERNEL_SIZE / KSLICES)  // 512
#define NGROUPS     (D_SIZE / 16)            // 128 row groups of 16

typedef float v2f __attribute__((ext_vector_type(2)));
typedef float v8f __attribute__((ext_vector_type(8)));

// ---------------------------------------------------------------------------
// CDNA5 async global->LDS staging (ASYNCcnt path), inline asm per ISA ch.10/15.18
// ---------------------------------------------------------------------------
__device__ __forceinline__ void async_load_b128_to_lds(uint32_t lds_byte_addr,
                                                       const float* gsrc) {
    // GV mode: vdst = LDS byte address, vaddr = 64-bit global address, saddr off
    asm volatile("global_load_async_to_lds_b128 %0, %1, off"
                 :
                 : "v"(lds_byte_addr), "v"((uint64_t)(uintptr_t)gsrc)
                 : "memory");
}

__device__ __forceinline__ void wait_asynccnt0() {
    asm volatile("s_wait_asynccnt 0" ::: "memory");
}

// ---------------------------------------------------------------------------
// K1a: partial row sums-of-squares via V_WMMA_F32_16X16X4_F32.
// D = A_sq x ones : every column of D equals the row-sum, so the result is
// invariant to K/N slot permutations; only the lane->M mapping matters.
// One wave handles (16 rows) x (512-sample k-slice); 4 accumulators for ILP.
// partials layout: [D_SIZE][KSLICES]
// ---------------------------------------------------------------------------
#if __has_builtin(__builtin_amdgcn_wmma_f32_16x16x4_f32)
#define HAVE_WMMA_F32 1
#else
#define HAVE_WMMA_F32 0
#endif

__global__ void k_norm_partials(const float* __restrict__ atoms,
                                float* __restrict__ partials) {
    const int tid  = threadIdx.x;
    const int lane = tid & 31;
    const int wave = tid >> 5;                      // 0..7
    const int task = blockIdx.x * 8 + wave;         // 0..1023
    const int group = task >> 3;                    // 0..127
    const int slice = task & (KSLICES - 1);         // 0..7
    const int rbase = group * 16;

#if HAVE_WMMA_F32
    const int half = lane >> 4;                     // 0: K=0,1   1: K=2,3
    const int sub  = lane & 15;                     // row within group
    const float* rowp = atoms + (size_t)(rbase + sub) * KERNEL_SIZE
                              + slice * KSLICE_LEN + half * 2;

    v8f acc0 = {}, acc1 = {}, acc2 = {}, acc3 = {};
    v2f ones; ones.x = 1.0f; ones.y = 1.0f;

    for (int k = 0; k < KSLICE_LEN; k += 16) {
        float2 x0 = *(const float2*)(rowp + k);
        float2 x1 = *(const float2*)(rowp + k + 4);
        float2 x2 = *(const float2*)(rowp + k + 8);
        float2 x3 = *(const float2*)(rowp + k + 12);
        v2f a0; a0.x = x0.x * x0.x; a0.y = x0.y * x0.y;
        v2f a1; a1.x = x1.x * x1.x; a1.y = x1.y * x1.y;
        v2f a2; a2.x = x2.x * x2.x; a2.y = x2.y * x2.y;
        v2f a3; a3.x = x3.x * x3.x; a3.y = x3.y * x3.y;
        acc0 = __builtin_amdgcn_wmma_f32_16x16x4_f32(false, a0, false, ones,
                                                     (short)0, acc0, false, false);
        acc1 = __builtin_amdgcn_wmma_f32_16x16x4_f32(false, a1, false, ones,
                                                     (short)0, acc1, false, false);
        acc2 = __builtin_amdgcn_wmma_f32_16x16x4_f32(false, a2, false, ones,
                                                     (short)0, acc2, false, false);
        acc3 = __builtin_amdgcn_wmma_f32_16x16x4_f32(false, a3, false, ones,
                                                     (short)0, acc3, false, false);
    }
    v8f acc = (acc0 + acc1) + (acc2 + acc3);

    // D layout: VGPR r holds M=r (lanes 0-15) and M=r+8 (lanes 16-31); any N.
    if (sub < 8) {
        float s = 0.0f;
        #pragma unroll
        for (int j = 0; j < 8; ++j)
            if (sub == j) s = acc[j];
        const int row = rbase + half * 8 + sub;
        partials[(size_t)row * KSLICES + slice] = s;
    }
#else
    // scalar fallback: lanes 0-15 each own one row of the group
    const int sub = lane & 15;
    if (lane < 16) {
        const float* rowp = atoms + (size_t)(rbase + sub) * KERNEL_SIZE
                                  + slice * KSLICE_LEN;
        float s = 0.0f;
        for (int k = 0; k < KSLICE_LEN; ++k) s += rowp[k] * rowp[k];
        partials[(size_t)(rbase + sub) * KSLICES + slice] = s;
    }
#endif
}

// ---------------------------------------------------------------------------
// K1b: inv_norm[d] = 1 / (sqrt(sum_slices) + 1e-8)
// ---------------------------------------------------------------------------
__global__ void k_finalize_norms(const float* __restrict__ partials,
                                 float* __restrict__ inv_norm) {
    const int d = blockIdx.x * blockDim.x + threadIdx.x;
    if (d < D_SIZE) {
        float s = 0.0f;
        #pragma unroll
        for (int j = 0; j < KSLICES; ++j) s += partials[(size_t)d * KSLICES + j];
        inv_norm[d] = 1.0f / (sqrtf(s) + 1e-8f);
    }
}

// ---------------------------------------------------------------------------
// K2: per (b,i): argmax over atom_selection row (2048) and positions row (512)
// First-occurrence tie-break to match jnp.argmax.
// ---------------------------------------------------------------------------
__global__ void k_argmax(const float* __restrict__ sel,   // (1024, 2048)
                         const float* __restrict__ pos,   // (1024, 512)
                         int* __restrict__ d_idx,         // (1024)
                         int* __restrict__ p_samp) {      // (1024) sample pos
    const int bi  = blockIdx.x;
    const int tid = threadIdx.x;
    __shared__ float sv[BLK];
    __shared__ int   si[BLK];

    // ---- argmax over D_SIZE ----
    {
        const float* row = sel + (size_t)bi * D_SIZE;
        float bv = -INFINITY; int bidx = 0x7fffffff;
        for (int j = tid; j < D_SIZE; j += BLK) {
            float v = row[j];
            if (v > bv) { bv = v; bidx = j; }   // ascending j keeps first max
        }
        sv[tid] = bv; si[tid] = bidx;
        __syncthreads();
        for (int st = BLK / 2; st > 0; st >>= 1) {
            if (tid < st) {
                float ov = sv[tid + st]; int oi = si[tid + st];
                if (ov > sv[tid] || (ov == sv[tid] && oi < si[tid])) {
                    sv[tid] = ov; si[tid] = oi;
                }
            }
            __syncthreads();
        }
        if (tid == 0) d_idx[bi] = si[0];
        __syncthreads();
    }

    // ---- argmax over N_FRAMES ----
    {
        const float* row = pos + (size_t)bi * N_FRAMES;
        float bv = -INFINITY; int bidx = 0x7fffffff;
        for (int j = tid; j < N_FRAMES; j += BLK) {
            float v = row[j];
            if (v > bv) { bv = v; bidx = j; }
        }
        sv[tid] = bv; si[tid] = bidx;
        __syncthreads();
        for (int st = BLK / 2; st > 0; st >>= 1) {
            if (tid < st) {
                float ov = sv[tid + st]; int oi = si[tid + st];
                if (ov > sv[tid] || (ov == sv[tid] && oi < si[tid])) {
                    sv[tid] = ov; si[tid] = oi;
                }
            }
            __syncthreads();
        }
        if (tid == 0) p_samp[bi] = si[0] * STEP;
    }
}

// ---------------------------------------------------------------------------
// K3: overlap-add. Block (b, c) owns output samples [c*TCHUNK, (c+1)*TCHUNK).
// For each of the 32 iterations (block-uniform), async-stage the overlapping
// atom segment into LDS, then accumulate amp * inv_norm * atom into registers.
// No atomics: each output sample written by exactly one block.
// ---------------------------------------------------------------------------
__global__ void k_overlap_add(const float* __restrict__ atoms,
                              const float* __restrict__ amps,     // (1024)
                              const float* __restrict__ inv_norm, // (2048)
                              const int*   __restrict__ d_idx,    // (1024)
                              const int*   __restrict__ p_samp,   // (1024)
                              float* __restrict__ out) {          // (32, 32768)
    const int b   = blockIdx.x;      // 0..31
    const int c   = blockIdx.y;      // 0..NCHUNK-1
    const int tid = threadIdx.x;
    const int t0  = c * TCHUNK;

    __shared__ __align__(16) float seg[TCHUNK];
    const uint32_t seg_base = (uint32_t)(uintptr_t)&seg[0];

    float acc[VPT];
    #pragma unroll
    for (int j = 0; j < VPT; ++j) acc[j] = 0.0f;

    for (int i = 0; i < N_ITERS; ++i) {
        const int bi = b * N_ITERS + i;
        const int p  = p_samp[bi];                       // multiple of 64
        int lo = t0 > p ? t0 : p;
        int hi = (t0 + TCHUNK < p + KERNEL_SIZE) ? (t0 + TCHUNK) : (p + KERNEL_SIZE);
        if (hi <= lo) continue;                          // block-uniform branch

        const int d   = d_idx[bi];
        const int k0  = lo - p;                          // multiple of 64
        const int len = hi - lo;                         // multiple of 64
        const float* gsrc = atoms + (size_t)d * KERNEL_SIZE + k0;  // 16B aligned

        // async stage [k0, k0+len) of row d into LDS (b128 per lane)
        const int nvec = len >> 2;                       // float4 count, mult of 16
        for (int v = tid; v < nvec; v += BLK) {
            async_load_b128_to_lds(seg_base + (uint32_t)(v << 4), gsrc + (v << 2));
        }
        wait_asynccnt0();
        __syncthreads();

        const float a = amps[bi] * inv_norm[d];
        #pragma unroll
        for (int j = 0; j < VPT; ++j) {
            const int t = t0 + tid + j * BLK;
            const int x = t - lo;                        // seg[x] = atoms[d, k0+x]
            if (x >= 0 && x < len) acc[j] += a * seg[x];
        }
        __syncthreads();                                 // before seg is re-staged
    }

    float* orow = out + (size_t)b * N_SAMPLES + t0;
    #pragma unroll
    for (int j = 0; j < VPT; ++j) orow[tid + j * BLK] = acc[j];
}

// ---------------------------------------------------------------------------
extern "C" void kernel_launch(void* const* d_in, const int* in_sizes, int n_in,
                              void* d_out, int out_size, void* d_ws, size_t ws_size,
                              hipStream_t stream) {
    const float* atom_selection = (const float*)d_in[0]; // (32, 32, 2048)
    const float* amps           = (const float*)d_in[1]; // (32, 32, 1)
    const float* positions      = (const float*)d_in[2]; // (32, 32, 512)
    const float* atoms          = (const float*)d_in[3]; // (2048, 4096)
    float* out = (float*)d_out;                          // (32, 1, 32768)

    // workspace layout
    char* ws = (char*)d_ws;
    float* inv_norm = (float*)(ws);                       // 2048 * 4 B
    int*   d_idx    = (int*)(ws + 8192);                  // 1024 * 4 B
    int*   p_samp   = (int*)(ws + 8192 + 4096);           // 1024 * 4 B
    float* partials = (float*)(ws + 16384);               // 2048 * 8 * 4 B

    k_norm_partials<<<dim3(NGROUPS * KSLICES / 8), dim3(BLK), 0, stream>>>(
        atoms, partials);
    k_finalize_norms<<<dim3(D_SIZE / BLK), dim3(BLK), 0, stream>>>(
        partials, inv_norm);
    k_argmax<<<dim3(BATCH * N_ITERS), dim3(BLK), 0, stream>>>(
        atom_selection, positions, d_idx, p_samp);
    k_overlap_add<<<dim3(BATCH, NCHUNK), dim3(BLK), 0, stream>>>(
        atoms, amps, inv_norm, d_idx, p_samp, out);
}